// MultiBoxLoss_2302102470943
// MI455X (gfx1250) — compile-verified
//
#include <hip/hip_runtime.h>
#include <math.h>

#define BB 64
#define PP 8732
#define OO 32
#define CC 81
#define IMGF 224.0f
#define THRESHF 0.5f
#define NTHREADS 512
#define NWAVES (NTHREADS / 32)
#define NCHUNKS ((PP + 15) / 16)   // 546 chunks of 16 priors

typedef __attribute__((ext_vector_type(2))) float v2f;
typedef __attribute__((ext_vector_type(8))) float v8f;

__device__ __forceinline__ float block_reduce_sum(float v, float* s_red, int tid) {
    s_red[tid] = v;
    __syncthreads();
    for (int s = NTHREADS / 2; s > 0; s >>= 1) {
        if (tid < s) s_red[tid] += s_red[tid + s];
        __syncthreads();
    }
    float r = s_red[0];
    __syncthreads();
    return r;
}

__global__ __launch_bounds__(NTHREADS)
void ssd_loss_per_image(const float* __restrict__ pred_locs,
                        const float* __restrict__ pred_scores,
                        const float* __restrict__ gt_boxes,
                        const int*   __restrict__ gt_labels,
                        const float* __restrict__ priors,
                        float*       __restrict__ partials /* [B][4] */) {
    __shared__ float  s_ce[PP];            // overlap_for_prior, then ce_neg
    __shared__ unsigned char s_obj[PP];    // object_for_prior (0..31)
    __shared__ unsigned char s_lab[PP];    // label_for_prior (0..80)
    __shared__ float4 s_bxy[OO];           // gt boxes xy (normalized)
    __shared__ float4 s_bc[OO];            // gt boxes cxcy
    __shared__ float  s_barea[OO];
    __shared__ int    s_blab[OO];
    __shared__ unsigned long long s_bestkey[OO];
    __shared__ float  s_red[NTHREADS];
    __shared__ int    s_cnt[33];
    __shared__ int    s_npos;
    __shared__ int    s_cntgt;

    const int img  = blockIdx.x;
    const int tid  = threadIdx.x;
    const int wave = tid >> 5;
    const int lane = tid & 31;

    // ---------------- Phase A: load GT boxes ----------------
    if (tid < OO) {
        float4 g = ((const float4*)gt_boxes)[img * OO + tid]; // [x,y,w,h] pixels
        float x0 = g.x / IMGF, y0 = g.y / IMGF;
        float x1 = (g.x + g.z) / IMGF, y1 = (g.y + g.w) / IMGF;
        s_bxy[tid]  = make_float4(x0, y0, x1, y1);
        s_bc[tid]   = make_float4((x0 + x1) * 0.5f, (y0 + y1) * 0.5f, x1 - x0, y1 - y0);
        s_barea[tid] = (x1 - x0) * (y1 - y0);
        s_blab[tid]  = gt_labels[img * OO + tid];
        s_bestkey[tid] = 0ull;
    }
    if (tid < 33) s_cnt[tid] = 0;
    if (tid == 0) { s_npos = 0; s_cntgt = 0; }
    __syncthreads();

    // ---------------- Phase B: per-prior argmax over objects ----------------
    for (int p = tid; p < PP; p += NTHREADS) {
        float4 pc = ((const float4*)priors)[p];
        float px0 = pc.x - pc.z * 0.5f, py0 = pc.y - pc.w * 0.5f;
        float px1 = pc.x + pc.z * 0.5f, py1 = pc.y + pc.w * 0.5f;
        float parea = (px1 - px0) * (py1 - py0);
        float best = -1.0f; int bo = 0;
        for (int o = 0; o < OO; ++o) {
            float4 b = s_bxy[o];
            float wx = fminf(b.z, px1) - fmaxf(b.x, px0);
            float wy = fminf(b.w, py1) - fmaxf(b.y, py0);
            wx = fmaxf(wx, 0.f); wy = fmaxf(wy, 0.f);
            float inter = wx * wy;
            float iou = inter / (s_barea[o] + parea - inter);
            if (iou > best) { best = iou; bo = o; }   // first-occurrence max
        }
        s_ce[p]  = best;
        s_obj[p] = (unsigned char)bo;
    }

    // ---------------- Phase B2: per-object argmax over priors (wave32 reduce) ----------------
    for (int o = wave; o < OO; o += NWAVES) {
        float4 b = s_bxy[o];
        float ba = s_barea[o];
        unsigned long long kbest = 0ull;
        for (int p = lane; p < PP; p += 32) {
            float4 pc = ((const float4*)priors)[p];
            float px0 = pc.x - pc.z * 0.5f, py0 = pc.y - pc.w * 0.5f;
            float px1 = pc.x + pc.z * 0.5f, py1 = pc.y + pc.w * 0.5f;
            float parea = (px1 - px0) * (py1 - py0);
            float wx = fmaxf(fminf(b.z, px1) - fmaxf(b.x, px0), 0.f);
            float wy = fmaxf(fminf(b.w, py1) - fmaxf(b.y, py0), 0.f);
            float inter = wx * wy;
            float iou = inter / (ba + parea - inter);
            // pack (iou, smallest-prior tiebreak): nonneg float bits are monotonic
            unsigned long long key =
                ((unsigned long long)__float_as_uint(iou) << 32) | (unsigned int)(~(unsigned int)p);
            if (key > kbest) kbest = key;
        }
        #pragma unroll
        for (int off = 16; off > 0; off >>= 1) {
            unsigned long long other = __shfl_xor(kbest, off);
            if (other > kbest) kbest = other;
        }
        if (lane == 0) s_bestkey[o] = kbest;
    }
    __syncthreads();

    // ---------------- Phase C: force-match (last object wins) ----------------
    if (tid == 0) {
        for (int o = 0; o < OO; ++o) {
            unsigned int lo = (unsigned int)(s_bestkey[o] & 0xffffffffull);
            int pstar = (int)(~lo);
            s_obj[pstar] = (unsigned char)o;
            s_ce[pstar]  = 1.0f;
        }
    }
    __syncthreads();

    // ---------------- Phase D: labels + SmoothL1 loc loss ----------------
    float loc_local = 0.f;
    for (int p = tid; p < PP; p += NTHREADS) {
        int o = s_obj[p];
        float ov = s_ce[p];
        int lbl = (ov < THRESHF) ? 0 : s_blab[o];
        s_lab[p] = (unsigned char)lbl;
        if (lbl != 0) {
            atomicAdd(&s_npos, 1);
            float4 bc = s_bc[o];
            float4 pc = ((const float4*)priors)[p];
            float t0 = (bc.x - pc.x) / (pc.z / 10.0f);
            float t1 = (bc.y - pc.y) / (pc.w / 10.0f);
            float t2 = logf(bc.z / pc.z) * 5.0f;
            float t3 = logf(bc.w / pc.w) * 5.0f;
            float4 pl = ((const float4*)pred_locs)[(size_t)img * PP + p];
            float d0 = fabsf(pl.x - t0), d1 = fabsf(pl.y - t1);
            float d2 = fabsf(pl.z - t2), d3 = fabsf(pl.w - t3);
            loc_local += (d0 < 1.f) ? 0.5f * d0 * d0 : d0 - 0.5f;
            loc_local += (d1 < 1.f) ? 0.5f * d1 * d1 : d1 - 0.5f;
            loc_local += (d2 < 1.f) ? 0.5f * d2 * d2 : d2 - 0.5f;
            loc_local += (d3 < 1.f) ? 0.5f * d3 * d3 : d3 - 0.5f;
        }
    }
    float locSum = block_reduce_sum(loc_local, s_red, tid);

    // ---------------- Phase E: cross entropy via WMMA row-sum reduction ----------------
    // Per wave: 16 priors x 84 classes (81 padded to 84). A-matrix (16x4 f32):
    // lane<16 -> row=lane, K={0,1}; lane>=16 -> row=lane-16, K={2,3}.
    const int m     = lane & 15;
    const int khalf = (lane >> 4) << 1;     // 0 or 2
    const v2f ones2 = {1.0f, 1.0f};
    float ce_local = 0.f;
    for (int chunk = wave; chunk < NCHUNKS; chunk += NWAVES) {
        int base = chunk * 16;
        int p = base + m;
        bool pv = p < PP;
        const float* sp = pred_scores + ((size_t)img * PP + (pv ? p : 0)) * CC;
        int pn = base + NWAVES * 16 + m;
        if (pn < PP)
            __builtin_prefetch(pred_scores + ((size_t)img * PP + pn) * CC, 0, 0);

        float vmax = -3.0e38f;
        v2f vals[21];
        #pragma unroll
        for (int j = 0; j < 21; ++j) {
            int c0 = 4 * j + khalf;
            float a = (pv && c0 < CC)       ? sp[c0]     : -3.0e38f;
            float b = (pv && (c0 + 1) < CC) ? sp[c0 + 1] : -3.0e38f;
            vals[j].x = a; vals[j].y = b;
            vmax = fmaxf(vmax, fmaxf(a, b));
        }
        vmax = fmaxf(vmax, __shfl_xor(vmax, 16));   // full row max (both K halves)

        v8f acc = {0.f, 0.f, 0.f, 0.f, 0.f, 0.f, 0.f, 0.f};
        #pragma unroll
        for (int j = 0; j < 21; ++j) {
            int c0 = 4 * j + khalf;
            v2f e;
            e.x = (pv && c0 < CC)       ? __expf(vals[j].x - vmax) : 0.f;
            e.y = (pv && (c0 + 1) < CC) ? __expf(vals[j].y - vmax) : 0.f;
            // D = E(16x4) * Ones(4x16) + D : every column = row sum of exp
            acc = __builtin_amdgcn_wmma_f32_16x16x4_f32(
                false, e, false, ones2, (short)0, acc, false, false);
        }
        // C/D layout: lane<16 VGPR q = D[q][lane]; lane>=16 VGPR q = D[8+q][lane-16]
        float mine = 0.f;
        #pragma unroll
        for (int q = 0; q < 8; ++q)
            if ((lane & 7) == q) mine = acc[q];
        float rowsum = __shfl(mine, m + 8);         // row m lives at lane m+8
        if (lane < 16 && pv) {
            float lse = logf(rowsum) + vmax;
            int lbl = s_lab[p];
            float ce = lse - sp[lbl];
            if (lbl != 0) { ce_local += ce; s_ce[p] = 0.f; }
            else          { s_ce[p] = fmaxf(ce, 0.f); }   // keep bits monotonic for select
        }
    }
    float cePos = block_reduce_sum(ce_local, s_red, tid);

    // ---------------- Phase F: exact top-k sum via radix select on float bits ----------------
    int npos = s_npos;
    int kk = 3 * ((npos > 0) ? npos : 1);
    if (kk > PP) kk = PP;

    unsigned int prefix = 0u;
    for (int b = 30; b >= 0; --b) {
        unsigned int trial = prefix | (1u << b);
        int cl = 0;
        for (int p = tid; p < PP; p += NTHREADS)
            if (__float_as_uint(s_ce[p]) >= trial) cl++;
        if (cl) atomicAdd(&s_cnt[b], cl);
        __syncthreads();
        if (s_cnt[b] >= kk) prefix = trial;
    }
    float T = __uint_as_float(prefix);   // kk-th largest value
    float sgt = 0.f; int cgt = 0;
    for (int p = tid; p < PP; p += NTHREADS) {
        float v = s_ce[p];
        if (__float_as_uint(v) > prefix) { sgt += v; cgt++; }
    }
    if (cgt) atomicAdd(&s_cntgt, cgt);
    float sumGT = block_reduce_sum(sgt, s_red, tid);

    if (tid == 0) {
        float hard = sumGT + (float)(kk - s_cntgt) * T;
        partials[img * 4 + 0] = (float)npos;
        partials[img * 4 + 1] = locSum;
        partials[img * 4 + 2] = cePos;
        partials[img * 4 + 3] = hard;
    }
}

__global__ void ssd_loss_finalize(const float* __restrict__ partials,
                                  float* __restrict__ out) {
    if (threadIdx.x == 0) {
        float tp = 0.f, ncl = 0.f, locs = 0.f, cepos = 0.f, hard = 0.f;
        for (int i = 0; i < BB; ++i) {
            float np = partials[i * 4 + 0];
            tp   += np;
            ncl  += fmaxf(np, 1.f);
            locs += partials[i * 4 + 1];
            cepos += partials[i * 4 + 2];
            hard += partials[i * 4 + 3];
        }
        float conf = (hard + cepos) / ncl;
        float loc  = (tp > 0.f) ? locs / fmaxf(tp * 4.f, 1.f) : 0.f;
        out[0] = conf + 1.0f * loc;   // ALPHA = 1
    }
}

extern "C" void kernel_launch(void* const* d_in, const int* in_sizes, int n_in,
                              void* d_out, int out_size, void* d_ws, size_t ws_size,
                              hipStream_t stream) {
    const float* pred_locs   = (const float*)d_in[0];
    const float* pred_scores = (const float*)d_in[1];
    const float* gt_boxes    = (const float*)d_in[2];
    const int*   gt_labels   = (const int*)d_in[3];
    const float* priors      = (const float*)d_in[4];
    float* partials = (float*)d_ws;   // 64 * 4 floats

    ssd_loss_per_image<<<BB, NTHREADS, 0, stream>>>(
        pred_locs, pred_scores, gt_boxes, gt_labels, priors, partials);
    ssd_loss_finalize<<<1, 32, 0, stream>>>(partials, (float*)d_out);
}